// MultiResHashGridMLP_41180146434628
// MI455X (gfx1250) — compile-verified
//
#include <hip/hip_runtime.h>
#include <math.h>
#include <stdint.h>

// Multi-resolution hash-grid encoding (Instant-NGP style) forward pass.
// 1M points x 16 levels x 8 corners; 8B random gathers from 64MB of tables
// (fully L2-resident on MI455X's 192MB L2). Gather-rate bound; WMMA does not
// apply. CDNA5 paths used: async global->LDS staging (ASYNCcnt), NT b128
// stores, packed f32 FMA accumulation, LDS store coalescing.

#define HG_LEVELS 16
#define HG_TPB 256
#define HG_OUT 35                 // 3 coords + 16 levels * 2 feats
#define HG_TBL_STRIDE (1u << 20)  // floats per level: 2^19 entries * 2 feats
#define HG_POW2_FROM 6            // levels >= 6 have hms == 2^19 (res^3 > 2^19)

typedef float v4f __attribute__((ext_vector_type(4)));
typedef float v2f __attribute__((ext_vector_type(2)));

struct HGParams {
  float    res[HG_LEVELS];
  uint32_t hms[HG_LEVELS];
  uint32_t magic[HG_LEVELS];      // floor(2^32 / hms)
};

#if defined(__HIP_DEVICE_COMPILE__) && defined(__gfx1250__) && \
    __has_builtin(__builtin_amdgcn_global_load_async_to_lds_b32)
#define HG_ASYNC 1
#else
#define HG_ASYNC 0
#endif

__global__ __launch_bounds__(HG_TPB) void hashgrid_fwd(
    const float* __restrict__ x,
    const float* __restrict__ tables,
    float* __restrict__ out,
    HGParams P, int npts)
{
  __shared__ __align__(16) float lds[HG_TPB * HG_OUT];  // 35840 B
  const int tid  = threadIdx.x;
  const int base = blockIdx.x * HG_TPB;

  // ---- Stage this block's 768 consecutive input floats into LDS ----
  {
    const int limf = (npts - base) * 3 < HG_TPB * 3 ? (npts - base) * 3
                                                    : HG_TPB * 3;
    const float* src = x + (size_t)base * 3;
#if HG_ASYNC
    for (int i = tid; i < limf; i += HG_TPB) {
      __builtin_amdgcn_global_load_async_to_lds_b32(
          (__attribute__((address_space(1))) int*)(src + i),
          (__attribute__((address_space(3))) int*)&lds[i], 0, 0);
    }
    asm volatile("s_wait_asynccnt 0" ::: "memory");
#else
    for (int i = tid; i < limf; i += HG_TPB) lds[i] = src[i];
#endif
  }
  __syncthreads();

  const bool active = (base + tid) < npts;
  const float px = lds[tid * 3 + 0];
  const float py = lds[tid * 3 + 1];
  const float pz = lds[tid * 3 + 2];
  __syncthreads();   // done reading the input staging region

  if (active) {
    lds[tid * HG_OUT + 0] = px;
    lds[tid * HG_OUT + 1] = py;
    lds[tid * HG_OUT + 2] = pz;

    #pragma unroll
    for (int l = 0; l < HG_LEVELS; ++l) {
      const float    r    = P.res[l];
      const uint32_t m    = P.hms[l];
      const uint32_t mask = m - 1u;       // exact for pow2 levels
      const uint32_t mg   = P.magic[l];
      const float* __restrict__ tbl = tables + (size_t)l * HG_TBL_STRIDE;

      const float sx = px * r, sy = py * r, sz = pz * r;
      const float fx = floorf(sx), fy = floorf(sy), fz = floorf(sz);
      const uint32_t ix = (uint32_t)(int)fx;
      const uint32_t iy = (uint32_t)(int)fy;
      const uint32_t iz = (uint32_t)(int)fz;
      const float tx = sx - fx, ty = sy - fy, tz = sz - fz;

      // Hash components for both corner choices per dim (primes 1, p1, p2).
      const uint32_t hx0 = ix;               const uint32_t hx1 = ix + 1u;
      const uint32_t hy0 = iy * 2654435761u; const uint32_t hy1 = (iy + 1u) * 2654435761u;
      const uint32_t hz0 = iz * 805459861u;  const uint32_t hz1 = (iz + 1u) * 805459861u;

      // Factored trilinear weights: 4 xy-products, combine z at the end.
      const float wx0 = 1.0f - tx, wy0 = 1.0f - ty, wz0 = 1.0f - tz;
      const float w00 = wx0 * wy0;   // corner bits (x=0,y=0)
      const float w10 = tx  * wy0;   // (x=1,y=0)
      const float w01 = wx0 * ty;    // (x=0,y=1)
      const float w11 = tx  * ty;    // (x=1,y=1)

      v2f a0 = {0.0f, 0.0f};         // z=0 plane accumulator (2 feats)
      v2f a1 = {0.0f, 0.0f};         // z=1 plane accumulator
      #pragma unroll
      for (int c = 0; c < 8; ++c) {
        // Corner bit set in dim d -> use xi+1 and weight xf (matches bin_mask).
        uint32_t h = ((c & 1) ? hx1 : hx0) ^ ((c & 2) ? hy1 : hy0)
                   ^ ((c & 4) ? hz1 : hz0);
        uint32_t hm;
        if (l >= HG_POW2_FROM) {
          hm = h & mask;                       // hms is 2^19: single v_and
        } else {
          // Exact h % m via magic multiply: q underestimates by at most 1.
          uint32_t q = __umulhi(h, mg);
          hm = h - q * m;
          if (hm >= m) hm -= m;
        }
        const v2f e = *(const v2f*)(tbl + 2u * hm);         // global_load_b64
        const float wxy = (c & 2) ? ((c & 1) ? w11 : w01)
                                  : ((c & 1) ? w10 : w00);  // folded: c is const
        if (c & 4) a1 += e * wxy;   // contract -> v_pk_fma_f32
        else       a0 += e * wxy;
      }
      const v2f a = a1 * tz + a0 * wz0;   // packed z-combine
      lds[tid * HG_OUT + 3 + 2 * l + 0] = a.x;  // stride-35 words: odd ->
      lds[tid * HG_OUT + 3 + 2 * l + 1] = a.y;  // bank-conflict-free (wave32)
    }
  }
  __syncthreads();

  // ---- Coalesced non-temporal writeback ----
  if (base + HG_TPB <= npts) {
    // Full block: 2240 b128 NT stores, fully coalesced.
    v4f* __restrict__ dst = (v4f*)(out + (size_t)base * HG_OUT);
    const v4f* s4 = (const v4f*)lds;
    for (int i = tid; i < (HG_TPB * HG_OUT) / 4; i += HG_TPB) {
      __builtin_nontemporal_store(s4[i], dst + i);
    }
  } else {
    const int limf = (npts - base) * HG_OUT;
    float* __restrict__ dst = out + (size_t)base * HG_OUT;
    for (int i = tid; i < limf; i += HG_TPB) {
      __builtin_nontemporal_store(lds[i], dst + i);
    }
  }
}

extern "C" void kernel_launch(void* const* d_in, const int* in_sizes, int n_in,
                              void* d_out, int out_size, void* d_ws, size_t ws_size,
                              hipStream_t stream) {
  (void)n_in; (void)out_size; (void)d_ws; (void)ws_size;

  // Level parameters with the same double-precision math as the reference.
  HGParams P;
  const double beta = exp((log(2048.0) - log(16.0)) / 15.0);
  for (int i = 0; i < HG_LEVELS; ++i) {
    const double r  = floor(16.0 * pow(beta, (double)i));
    const double h3 = r * r * r;
    const double hm = h3 < 524288.0 ? h3 : 524288.0;   // min(res^3, 2^19)
    P.res[i]   = (float)r;
    P.hms[i]   = (uint32_t)hm;
    P.magic[i] = (uint32_t)(0x100000000ULL / (uint64_t)P.hms[i]);
  }

  const float* x      = (const float*)d_in[0];
  const float* tables = (const float*)d_in[1];
  float*       outp   = (float*)d_out;
  const int npts   = in_sizes[0] / 3;                  // 1048576
  const int blocks = (npts + HG_TPB - 1) / HG_TPB;     // 4096

  hipLaunchKernelGGL(hashgrid_fwd, dim3(blocks), dim3(HG_TPB), 0, stream,
                     x, tables, outp, P, npts);
}